// GINConv_70935679861205
// MI455X (gfx1250) — compile-verified
//
#include <hip/hip_runtime.h>

// Fused GINConv + Linear + LayerNorm for gfx1250 (MI455X).
// One block = 256 threads = 8 wave32, handles MT=32 node rows.
//   Stage 1: gather-sum neighbors (L2-resident) + eps-residual -> h in LDS
//   Stage 2: W^T staged in LDS (padded stride, bank-conflict free)
//   Stage 3: y = h @ W via V_WMMA_F32_16X16X4_F32 (wave j owns cols 16j..16j+15,
//            two 16-row tiles -> two v8f accumulators)
//   Stage 4: + bias, LayerNorm via LDS + wave32 shuffle reduction, store.

#define C_DIM 128
#define MT    32          // rows per block
#define HS    132         // LDS row stride in floats (132 % 64 == 4 -> no bank conflicts)

typedef __attribute__((ext_vector_type(2))) float v2f;
typedef __attribute__((ext_vector_type(4))) float v4f;
typedef __attribute__((ext_vector_type(8))) float v8f;

// Only check for the target builtin in the device pass; it is (correctly)
// absent during the x86 host pass of the HIP TU.
#if defined(__HIP_DEVICE_COMPILE__) && !__has_builtin(__builtin_amdgcn_wmma_f32_16x16x4_f32)
#error "missing __builtin_amdgcn_wmma_f32_16x16x4_f32 on this toolchain (device pass)"
#endif

__global__ __launch_bounds__(256) void gin_fused_kernel(
    const float* __restrict__ x,          // [N, 128]
    const long long* __restrict__ edge,   // [N, K] int64, value N == zero pad
    const float* __restrict__ epsp,       // [1]
    const float* __restrict__ W,          // [128, 128] row-major (C_in x C_out)
    const float* __restrict__ bias,       // [128]
    const float* __restrict__ gamma,      // [128]
    const float* __restrict__ beta,       // [128]
    float* __restrict__ out,              // [N, 128]
    int N, int K)
{
    __shared__ float w_lds[C_DIM * HS];   // W transposed: w_lds[n*HS + k] = W[k][n]
    __shared__ float h_lds[MT * HS];      // h tile, later reused for y tile

    const int tid  = threadIdx.x;
    const int lane = tid & 31;
    const int wv   = tid >> 5;            // 0..7 : wave id = output column tile
    const int l16  = lane & 15;
    const int half = lane >> 4;           // 0 or 1

    const int rowBase = blockIdx.x * MT;
    const float eps1 = 1.0f + epsp[0];

    // ---- Stage 2 (issue early): load W transposed into LDS -------------------
    for (int idx = tid; idx < C_DIM * C_DIM; idx += 256) {
        const int k = idx >> 7;           // row of W  (C_in)
        const int n = idx & 127;          // col of W  (C_out)
        w_lds[n * HS + k] = W[idx];       // coalesced global read
    }

    // ---- Stage 1: gather + eps-residual, each wave owns 4 rows ---------------
    for (int i = 0; i < 4; ++i) {
        const int lrow = wv * 4 + i;              // 0..31
        const int row  = rowBase + lrow;
        v4f acc = {0.0f, 0.0f, 0.0f, 0.0f};
        if (row < N) {
            const v4f xv = *(const v4f*)(x + (size_t)row * C_DIM + lane * 4);
            acc = eps1 * xv;
            const long long* er = edge + (size_t)row * K;
            for (int k = 0; k < K; ++k) {
                const long long nb = er[k];       // wave-uniform value
                if (nb < (long long)N) {
                    acc += *(const v4f*)(x + (size_t)nb * C_DIM + lane * 4);
                }
            }
        }
        *(v4f*)(h_lds + lrow * HS + lane * 4) = acc;   // zero rows past N
    }
    __syncthreads();

    // ---- Stage 3: y = h @ W with f32 WMMA (16x16x4), 32 K-steps --------------
    v8f c0 = {};   // rows 0..15  of tile, cols 16*wv..16*wv+15
    v8f c1 = {};   // rows 16..31 of tile
    const float* wcol = w_lds + (wv * 16 + l16) * HS;  // this lane's W column
    #pragma unroll 4
    for (int s = 0; s < C_DIM / 4; ++s) {
        const int k0 = 4 * s + 2 * half;  // even -> 8B-aligned v2f loads
        const v2f b  = *(const v2f*)(wcol + k0);
        const v2f a0 = *(const v2f*)(h_lds + l16 * HS + k0);
        const v2f a1 = *(const v2f*)(h_lds + (l16 + 16) * HS + k0);
        c0 = __builtin_amdgcn_wmma_f32_16x16x4_f32(false, a0, false, b,
                                                   (short)0, c0, false, false);
        c1 = __builtin_amdgcn_wmma_f32_16x16x4_f32(false, a1, false, b,
                                                   (short)0, c1, false, false);
    }
    __syncthreads();   // everyone done reading h_lds before we overwrite with y

    // ---- Stage 4a: spill y (+bias) to LDS ------------------------------------
    const int ncol = wv * 16 + l16;
    const float bn = bias[ncol];
    #pragma unroll
    for (int r = 0; r < 8; ++r) {
        const int m0 = r + 8 * half;                       // 0..15
        h_lds[m0 * HS + ncol]        = c0[r] + bn;
        h_lds[(m0 + 16) * HS + ncol] = c1[r] + bn;
    }
    __syncthreads();

    // ---- Stage 4b: LayerNorm, each wave owns 4 rows, lane covers 4 channels --
    const v4f g  = *(const v4f*)(gamma + lane * 4);
    const v4f be = *(const v4f*)(beta  + lane * 4);
    for (int i = 0; i < 4; ++i) {
        const int lrow = wv * 4 + i;
        const int row  = rowBase + lrow;
        const v4f yv = *(const v4f*)(h_lds + lrow * HS + lane * 4);
        float s  = yv.x + yv.y + yv.z + yv.w;
        float sq = yv.x * yv.x + yv.y * yv.y + yv.z * yv.z + yv.w * yv.w;
        #pragma unroll
        for (int off = 16; off >= 1; off >>= 1) {          // wave32 reduction
            s  += __shfl_xor(s,  off, 32);
            sq += __shfl_xor(sq, off, 32);
        }
        const float mu   = s * (1.0f / 128.0f);
        const float var  = sq * (1.0f / 128.0f) - mu * mu;
        const float rstd = rsqrtf(var + 1e-5f);
        v4f o;
        o.x = (yv.x - mu) * rstd * g.x + be.x;
        o.y = (yv.y - mu) * rstd * g.y + be.y;
        o.z = (yv.z - mu) * rstd * g.z + be.z;
        o.w = (yv.w - mu) * rstd * g.w + be.w;
        if (row < N) {
            *(v4f*)(out + (size_t)row * C_DIM + lane * 4) = o;
        }
    }
}

extern "C" void kernel_launch(void* const* d_in, const int* in_sizes, int n_in,
                              void* d_out, int out_size, void* d_ws, size_t ws_size,
                              hipStream_t stream) {
    const float*     x     = (const float*)d_in[0];
    const long long* edge  = (const long long*)d_in[1];   // int64 in reference
    const float*     eps   = (const float*)d_in[2];
    const float*     W     = (const float*)d_in[3];
    const float*     bias  = (const float*)d_in[4];
    const float*     gamma = (const float*)d_in[5];
    const float*     beta  = (const float*)d_in[6];
    float*           out   = (float*)d_out;

    const int N = in_sizes[0] / C_DIM;       // 50000
    const int K = in_sizes[1] / N;           // 32

    const int grid = (N + MT - 1) / MT;      // 1563 blocks
    gin_fused_kernel<<<grid, 256, 0, stream>>>(x, edge, eps, W, bias, gamma, beta,
                                               out, N, K);
}